// CtoX_58016418234799
// MI455X (gfx1250) — compile-verified
//
#include <hip/hip_runtime.h>

typedef __attribute__((ext_vector_type(2))) float v2f;
typedef __attribute__((ext_vector_type(4))) float v4f;
typedef __attribute__((ext_vector_type(8))) float v8f;

#define EPSF 1e-8f
#define BIGF 100000.0f

static __device__ __forceinline__ v4f v4min(v4f a, v4f b) {
  v4f r;
  r.x = fminf(a.x, b.x); r.y = fminf(a.y, b.y);
  r.z = fminf(a.z, b.z); r.w = fminf(a.w, b.w);
  return r;
}
static __device__ __forceinline__ v4f v4max(v4f a, v4f b) {
  v4f r;
  r.x = fmaxf(a.x, b.x); r.y = fmaxf(a.y, b.y);
  r.z = fmaxf(a.z, b.z); r.w = fmaxf(a.w, b.w);
  return r;
}

// ---------------------------------------------------------------------------
// Stage 1: per (b, ni, nj) 16x16 atom block, reduce 256 atom pairs x 64 ch
// -> z1 row (256 features: mean | min | max | std).
// Thread (p, cg): fixed atom-row p (16 of them), 4 channels (float4 loads),
// loops q=0..15. Single pass: sum, mask*sum, mask*sum^2, min/max with BIG
// offsets. mask(p,q) = a_p*b_q factorizes -> count = (sum a)(sum b).
// 16 p-partials combined with a 4-step LDS tree.
// ---------------------------------------------------------------------------
__global__ __launch_bounds__(256) void stats1_kernel(
    const float* __restrict__ delta1,   // (4,256,256,64)
    const float* __restrict__ cmask1,   // (4,256)
    const float* __restrict__ cmask2,   // (4,256)
    float* __restrict__ z1)             // (1024,256)
{
  const int t  = threadIdx.x;
  const int cg = t & 15;               // channel group: channels 4*cg..4*cg+3
  const int p  = t >> 4;               // atom row within block: 0..15
  const int blk = blockIdx.x;          // 0..1023 = b*256 + ni*16 + nj
  const int b  = blk >> 8;
  const int ni = (blk >> 4) & 15;
  const int nj = blk & 15;

  __shared__ float sa[16], sb[16];
  __shared__ v4f red[16][5][16];       // [p][stat][cg], 20 KB

  if (t < 16)      sa[t]      = cmask1[b*256 + ni*16 + t];
  else if (t < 32) sb[t - 16] = cmask2[b*256 + nj*16 + (t - 16)];
  __syncthreads();

  const float ap = sa[p];
  v4f s   = {0.f, 0.f, 0.f, 0.f};
  v4f ms  = s, ms2 = s;
  v4f mn  = { 3.4e38f,  3.4e38f,  3.4e38f,  3.4e38f};
  v4f mx  = {-3.4e38f, -3.4e38f, -3.4e38f, -3.4e38f};

  const float* base = delta1 +
      (((size_t)b * 256 + (size_t)ni * 16 + p) * 256 + (size_t)nj * 16) * 64 + cg * 4;

  #pragma unroll
  for (int q = 0; q < 16; ++q) {
    const v4f v = *(const v4f*)(base + q * 64);
    const float mpq = ap * sb[q];              // amortized over 4 channels
    const float off = BIGF * (1.f - mpq);
    s   += v;
    const v4f mv = mpq * v;
    ms  += mv;
    ms2 += mv * v;
    mn = v4min(mn, v + off);
    mx = v4max(mx, v - off);
  }

  red[p][0][cg] = s;  red[p][1][cg] = ms; red[p][2][cg] = ms2;
  red[p][3][cg] = mn; red[p][4][cg] = mx;
  __syncthreads();

  // tree reduce over p
  #pragma unroll
  for (int step = 8; step >= 1; step >>= 1) {
    if (p < step) {
      red[p][0][cg] += red[p + step][0][cg];
      red[p][1][cg] += red[p + step][1][cg];
      red[p][2][cg] += red[p + step][2][cg];
      red[p][3][cg]  = v4min(red[p][3][cg], red[p + step][3][cg]);
      red[p][4][cg]  = v4max(red[p][4][cg], red[p + step][4][cg]);
    }
    __syncthreads();
  }

  if (p == 0) {
    s = red[0][0][cg]; ms = red[0][1][cg]; ms2 = red[0][2][cg];
    mn = red[0][3][cg]; mx = red[0][4][cg];
    float asum = 0.f, bsum = 0.f;
    #pragma unroll
    for (int i = 0; i < 16; ++i) { asum += sa[i]; bsum += sb[i]; }
    const float cnt    = asum * bsum;
    const float divide = cnt + EPSF;
    const v4f   m      = s / divide;
    const v4f   stdv   = (ms2 - 2.f * m * ms + m * m * cnt) / divide;
    float* zrow = z1 + (size_t)blk * 256;
    *(v4f*)(zrow + cg * 4)       = m;
    *(v4f*)(zrow + 64 + cg * 4)  = mn;
    *(v4f*)(zrow + 128 + cg * 4) = mx;
    *(v4f*)(zrow + 192 + cg * 4) = stdv;
  }
}

// ---------------------------------------------------------------------------
// Stage 2 stats: reduce E (1024 x 64, rows = b*256 + i*16 + j) over j per
// (b,i) -> z2 row (256 features).
// ---------------------------------------------------------------------------
__global__ __launch_bounds__(64) void stats2_kernel(
    const float* __restrict__ E,        // (1024, 64)
    const float* __restrict__ emask2,   // (4, 16)
    float* __restrict__ z2)             // (64, 256)
{
  const int c   = threadIdx.x;
  const int blk = blockIdx.x;           // 0..63 = b*16 + i
  const int b = blk >> 4, i = blk & 15;

  __shared__ float em[16];
  if (c < 16) em[c] = emask2[b * 16 + c];
  __syncthreads();

  float s = 0.f, ms = 0.f, ms2 = 0.f, denom = 0.f;
  float mn = 3.4e38f, mx = -3.4e38f;
  const float* base = E + ((size_t)(b * 256 + i * 16)) * 64 + c;
  #pragma unroll
  for (int j = 0; j < 16; ++j) {
    const float v = base[j * 64];
    const float e = em[j];
    denom += e;
    s += v; ms += e * v; ms2 += e * v * v;
    const float off = BIGF * (1.f - e);
    mn = fminf(mn, v + off);
    mx = fmaxf(mx, v - off);
  }
  const float m    = s / denom;                              // ref has no EPS here
  const float stdv = (ms2 - 2.f * m * ms + m * m * denom) / denom;
  float* zr = z2 + (size_t)blk * 256;
  zr[c]       = m;
  zr[64 + c]  = mn;
  zr[128 + c] = mx;
  zr[192 + c] = stdv;
}

// ---------------------------------------------------------------------------
// WMMA GEMM: D(M,N) = A(M,K) * W(N,K)^T + bias(N)   [fp32, V_WMMA_F32_16X16X4_F32]
// One wave (32 lanes, EXEC all ones) per 16x16 output tile; K stepped by 4.
// A-layout (16x4 fp32): lane L -> row L&15, K-pair base 2*(L>>4).
// B = W^T, W row-major (N,K): B[k][n] = W[n*K + k] -> same lane addressing as A.
// C/D (16x16 fp32): acc[r] = D[row = r + 8*(L>>4)][col = L&15] within tile.
// ---------------------------------------------------------------------------
__global__ __launch_bounds__(32) void wmma_gemm_kernel(
    const float* __restrict__ A, const float* __restrict__ W,
    const float* __restrict__ bias, float* __restrict__ D,
    int Ntiles, int K, int N)
{
  const int tile = blockIdx.x;
  const int tn = tile % Ntiles;
  const int tm = tile / Ntiles;
  const int lane = threadIdx.x;        // 0..31
  const int l16  = lane & 15;
  const int hi   = lane >> 4;          // 0 or 1 -> K offset 0 or 2

  const float* arow = A + (size_t)(tm * 16 + l16) * K + 2 * hi;
  const float* wrow = W + (size_t)(tn * 16 + l16) * K + 2 * hi;

  v8f acc = {};
  for (int k = 0; k < K; k += 4) {
    v2f a  = { arow[k], arow[k + 1] };
    v2f bv = { wrow[k], wrow[k + 1] };
    acc = __builtin_amdgcn_wmma_f32_16x16x4_f32(
        /*neg_a=*/false, a, /*neg_b=*/false, bv,
        /*c_mod=*/(short)0, acc, /*reuse_a=*/false, /*reuse_b=*/false);
  }

  const float bb = bias[tn * 16 + l16];
  float* drow = D + (size_t)(tm * 16 + 8 * hi) * N + tn * 16 + l16;
  #pragma unroll
  for (int r = 0; r < 8; ++r)
    drow[(size_t)r * N] = acc[r] + bb;
}

// ---------------------------------------------------------------------------
extern "C" void kernel_launch(void* const* d_in, const int* in_sizes, int n_in,
                              void* d_out, int out_size, void* d_ws, size_t ws_size,
                              hipStream_t stream) {
  const float* delta1  = (const float*)d_in[0];   // (4,256,256,64)
  const float* c_mask1 = (const float*)d_in[1];   // (4,256,1,1)
  const float* c_mask2 = (const float*)d_in[2];   // (4,1,256,1)
  const float* e_mask2 = (const float*)d_in[3];   // (4,1,16,1)
  const float* W1      = (const float*)d_in[4];   // (64, 256)
  const float* b1      = (const float*)d_in[5];   // (64)
  const float* W2      = (const float*)d_in[6];   // (256, 256)
  const float* b2      = (const float*)d_in[7];   // (256)

  float* z1 = (float*)d_ws;                       // 1024*256
  float* E  = z1 + 1024 * 256;                    // 1024*64
  float* z2 = E + 1024 * 64;                      // 64*256

  // Stage 1 stats: one block per (b, ni, nj) tile.  (HBM-bound: 67 MB read)
  stats1_kernel<<<1024, 256, 0, stream>>>(delta1, c_mask1, c_mask2, z1);

  // GEMM1: (1024 x 256) x (256 x 64) + b1 -> E.  64 M-tiles x 4 N-tiles.
  wmma_gemm_kernel<<<64 * 4, 32, 0, stream>>>(z1, W1, b1, E, /*Ntiles=*/4,
                                              /*K=*/256, /*N=*/64);

  // Stage 2 stats: one block per (b, i).
  stats2_kernel<<<64, 64, 0, stream>>>(E, e_mask2, z2);

  // GEMM2: (64 x 256) x (256 x 256) + b2 -> out.  4 M-tiles x 16 N-tiles.
  wmma_gemm_kernel<<<4 * 16, 32, 0, stream>>>(z2, W2, b2, (float*)d_out,
                                              /*Ntiles=*/16, /*K=*/256, /*N=*/256);
}